// EdgeGRU_74955769249851
// MI455X (gfx1250) — compile-verified
//
#include <hip/hip_runtime.h>
#include <cstdint>

#define WSTEPS 24
#define ROWPAD 25           // 24 + 1 pad; 25 coprime with 64 banks -> conflict-free LDS reads
#define BLOCK  256
#define F4_PER_ROW 6        // 24 floats = 6 float4
#define F4_PER_TILE (BLOCK * F4_PER_ROW)

__device__ __forceinline__ float fast_exp2(float x) { return __builtin_amdgcn_exp2f(x); }
__device__ __forceinline__ float fast_log2(float x) { return __builtin_amdgcn_logf(x); }
__device__ __forceinline__ float fast_rcp(float x)  { return __builtin_amdgcn_rcpf(x); }

__device__ __forceinline__ float fast_sigmoid(float x) {
    // sigmoid(x) = 1 / (1 + 2^(-x*log2e)) : one v_exp_f32 + one v_rcp_f32
    return fast_rcp(1.0f + fast_exp2(-1.4426950408889634f * x));
}

__device__ __forceinline__ float fast_tanh(float x) {
#if __has_builtin(__builtin_amdgcn_tanhf)
    return __builtin_amdgcn_tanhf(x);        // v_tanh_f32 (CDNA5 trans op)
#else
    // tanh(x) = 2*sigmoid(2x) - 1
    return __builtin_fmaf(2.0f, fast_sigmoid(2.0f * x), -1.0f);
#endif
}

__global__ __launch_bounds__(BLOCK) void edge_gru_gfx1250(
    const float* __restrict__ alpha,   // [E, 24] row-major
    const float* __restrict__ w_ih,    // [3]
    const float* __restrict__ w_hh,    // [3]
    const float* __restrict__ b_ih,    // [3]
    const float* __restrict__ b_hh,    // [3]
    float* __restrict__ out,           // [E]
    int E)
{
    __shared__ float tile[BLOCK * ROWPAD];   // 25,600 B of the 320 KB WGP LDS

    const int tid = threadIdx.x;
    const int e0  = blockIdx.x * BLOCK;

    // LDS byte offset of 'tile' (flat shared ptr: low 32 bits == LDS offset)
    const unsigned lds_base = (unsigned)(uintptr_t)(&tile[0]);

    // ---- Stage tile: coalesced async copies global -> LDS (ASYNCcnt path) ----
    // 1536 float4s per tile, 256 lanes -> 6 iterations; every lane issues one
    // global_load_async_to_lds_b128 with a unique LDS destination.
#pragma unroll
    for (int it = 0; it < F4_PER_ROW; ++it) {
        const int flat = it * BLOCK + tid;        // 0..1535
        const int er   = flat / F4_PER_ROW;       // edge within tile: 0..255
        const int k4   = flat % F4_PER_ROW;       // float4 index within row: 0..5
        int ge = e0 + er;
        if (ge > E - 1) ge = E - 1;               // clamp source row (dest stays unique)
        const uint64_t gaddr = (uint64_t)(alpha + ((size_t)ge * WSTEPS + (size_t)(k4 * 4)));
        const unsigned laddr = lds_base + (unsigned)(er * (ROWPAD * 4) + k4 * 16);
        asm volatile("global_load_async_to_lds_b128 %0, %1, off"
                     :: "v"(laddr), "v"(gaddr)
                     : "memory");
    }
    // Inline-asm async ops are invisible to compiler waitcnt insertion:
    asm volatile("s_wait_asynccnt 0x0" ::: "memory");
    __syncthreads();

    // ---- Scalar GRU weights (uniform -> scalar loads) ----
    const float wi0 = w_ih[0], wi1 = w_ih[1], wi2 = w_ih[2];
    const float wh0 = w_hh[0], wh1 = w_hh[1], wh2 = w_hh[2];
    const float bi2 = b_ih[2];
    const float br  = b_ih[0] + b_hh[0];
    const float bz  = b_ih[1] + b_hh[1];
    const float bh2 = b_hh[2];

    // ---- 24-step recurrence; x_t from LDS (pitch 25 -> bank-conflict-free) ----
    const float* row = &tile[tid * ROWPAD];
    float h = 0.0f;
#pragma unroll
    for (int t = 0; t < WSTEPS; ++t) {
        const float x = row[t];
        const float r = fast_sigmoid(__builtin_fmaf(wi0, x, __builtin_fmaf(wh0, h, br)));
        const float z = fast_sigmoid(__builtin_fmaf(wi1, x, __builtin_fmaf(wh1, h, bz)));
        const float n = fast_tanh(__builtin_fmaf(wi2, x, bi2) +
                                  r * __builtin_fmaf(wh2, h, bh2));
        h = __builtin_fmaf(z, h - n, n);          // (1-z)*n + z*h
    }

    // ---- softplus(h); |h| <= 1 so the direct formula is numerically safe ----
    const int ge = e0 + tid;
    if (ge < E) {
        const float sp = 0.6931471805599453f *
                         fast_log2(1.0f + fast_exp2(1.4426950408889634f * h));
        out[ge] = sp;
    }
}

extern "C" void kernel_launch(void* const* d_in, const int* in_sizes, int n_in,
                              void* d_out, int out_size, void* d_ws, size_t ws_size,
                              hipStream_t stream) {
    (void)n_in; (void)d_ws; (void)ws_size; (void)out_size;
    const float* alpha = (const float*)d_in[0];
    const float* w_ih  = (const float*)d_in[1];
    const float* w_hh  = (const float*)d_in[2];
    const float* b_ih  = (const float*)d_in[3];
    const float* b_hh  = (const float*)d_in[4];
    float* out = (float*)d_out;

    const int E = in_sizes[0] / WSTEPS;
    const int blocks = (E + BLOCK - 1) / BLOCK;
    hipLaunchKernelGGL(edge_gru_gfx1250, dim3(blocks), dim3(BLOCK), 0, stream,
                       alpha, w_ih, w_hh, b_ih, b_hh, out, E);
}